// PairwiseComparison_32358283608329
// MI455X (gfx1250) — compile-verified
//
#include <hip/hip_runtime.h>

typedef float v2f __attribute__((ext_vector_type(2)));
typedef float v8f __attribute__((ext_vector_type(8)));

#define BSZ 16
#define SZ  512
#define NF  32

// dp[b,f,i,j] = relu(w0[f]*t[b,i] + w1[f]*t[b,j] + w2[f])
// via V_WMMA_F32_16X16X4_F32:  D = A x B,  A row m = [a_m, 1, 0, 0],
// B col n = [1, c_n, 0, 0]^T  =>  D[m][n] = a_m + c_n   (fp32 exact)
__global__ __launch_bounds__(256) void pairwise_wmma_kernel(
    const float* __restrict__ t,   // (BSZ, SZ)
    const float* __restrict__ w,   // (1, NF, 3, 1) -> flat [f*3 + k]
    float* __restrict__ out)       // (BSZ, NF, SZ*SZ)
{
    const int lane  = threadIdx.x & 31;
    const int wave  = threadIdx.x >> 5;
    const int laneN = lane & 15;
    const bool lo   = lane < 16;

    // block -> (b, f, i-tile); 32 i-tiles, 32 features, 16 batches
    const int bid   = blockIdx.x;
    const int itile = bid & 31;
    const int f     = (bid >> 5) & (NF - 1);
    const int b     = bid >> 10;

    const float w0 = w[f * 3 + 0];
    const float w1 = w[f * 3 + 1];
    const float w2 = w[f * 3 + 2];

    const int i0 = itile * 16;

    // A matrix (16x4 f32): lane<16 holds row M=laneN, {K0,K1} = {a_M, 1};
    // lanes 16-31 hold {K2,K3} = {0,0}. a_i = fma(w0, t[b,i], w2).
    const float ti = t[b * SZ + i0 + laneN];   // valid address for all lanes
    const float a  = fmaf(w0, ti, w2);
    v2f A;
    A.x = lo ? a    : 0.0f;
    A.y = lo ? 1.0f : 0.0f;

    // Each wave owns 64 consecutive columns = 4 WMMA tiles.
    const int jbase = wave * 64;
    float* const slice = out + (((size_t)(b * NF + f)) << 18) + (size_t)i0 * SZ;
    // Lane's store base: row offset 0 (lanes 0-15) or 8 (lanes 16-31), col laneN.
    float* const pbase = slice + (size_t)(lo ? 0 : 8) * SZ + laneN;

    const float pinf = __builtin_inff();

    #pragma unroll
    for (int jt = 0; jt < 4; ++jt) {
        const int j0 = jbase + jt * 16;
        const float cj = w1 * t[b * SZ + j0 + laneN];

        // B matrix (4x16 f32), layout-hedged: ones row at V0/lanes0-15 (K0),
        // c placed at BOTH V1/lanes0-15 and V0/lanes16-31 so that whichever
        // of those slots is K1 contributes 1*c; the other slot maps to a
        // K-position where A is 0.
        v2f B;
        B.x = lo ? 1.0f : cj;
        B.y = lo ? cj   : 0.0f;

        v8f acc = {};
        acc = __builtin_amdgcn_wmma_f32_16x16x4_f32(
            /*neg_a=*/false, A, /*neg_b=*/false, B,
            /*c_mod=*/(short)0, acc, /*reuse_a=*/false, /*reuse_b=*/false);

        // ReLU via single v_med3_f32(x, 0, +inf), then store.
        // Accumulator VGPR k = output row (k | hi8), col laneN; each b32
        // store instruction writes two dense 64B row segments.
        float* p = pbase + j0;
        #pragma unroll
        for (int k = 0; k < 8; ++k) {
            p[(size_t)k * SZ] = __builtin_amdgcn_fmed3f(acc[k], 0.0f, pinf);
        }
    }
}

extern "C" void kernel_launch(void* const* d_in, const int* in_sizes, int n_in,
                              void* d_out, int out_size, void* d_ws, size_t ws_size,
                              hipStream_t stream) {
    const float* t = (const float*)d_in[0];   // (16, 512) fp32
    const float* w = (const float*)d_in[1];   // (1, 32, 3, 1) fp32
    float* out = (float*)d_out;               // 16*32*512*512 fp32

    dim3 grid(BSZ * NF * (SZ / 16));          // 16384 blocks
    dim3 block(256);                          // 8 wave32
    pairwise_wmma_kernel<<<grid, block, 0, stream>>>(t, w, out);
}